// GConvLSTMCell_43258910605774
// MI455X (gfx1250) — compile-verified
//
#include <hip/hip_runtime.h>
#include <cstdint>
#include <cstddef>

typedef __attribute__((ext_vector_type(2))) float v2f;
typedef __attribute__((ext_vector_type(8))) float v8f;

namespace {
constexpr int kB   = 8;
constexpr int kN   = 10000;
constexpr int kCIN = 64;
constexpr int kH   = 64;
constexpr int kE   = 160000;
constexpr int kK   = kCIN + kH;   // 128
constexpr int kF   = 4 * kH;      // 256
constexpr long long kBN  = (long long)kB * kN;   // 80000
constexpr long long kBNF = kBN * kF;             // 20,480,000
constexpr long long kBNH = kBN * kH;             // 5,120,000
}

// ---------- gcn_norm pieces ----------
__global__ void k_deg_init(float* __restrict__ deg) {
  int n = blockIdx.x * blockDim.x + threadIdx.x;
  if (n < kN) deg[n] = 1.0f;   // self-loop weight contributes 1 to every node
}

__global__ void k_deg_accum(const long long* __restrict__ ei,
                            const float* __restrict__ ew,
                            float* __restrict__ deg) {
  int e = blockIdx.x * blockDim.x + threadIdx.x;
  if (e < kE) {
    long long d = ei[(long long)kE + e];   // dst row of edge_index
    atomicAdd(&deg[d], ew[e]);
  }
}

__global__ void k_deg_rsqrt(float* __restrict__ deg) {
  int n = blockIdx.x * blockDim.x + threadIdx.x;
  if (n < kN) {
    float d = deg[n];
    deg[n] = (d > 0.0f) ? rsqrtf(fmaxf(d, 1e-12f)) : 0.0f;
  }
}

__global__ void k_zero(float* __restrict__ p, long long count) {
  long long i = (long long)blockIdx.x * blockDim.x + threadIdx.x;
  if (i < count) p[i] = 0.0f;
}

// ---------- dense GEMM: xw = concat(x,h) @ W  via V_WMMA_F32_16X16X4_F32 ----------
// One wave per 16x16 output tile. M = 80000 (5000 tiles), Nout = 256 (16 tiles), K = 128.
__global__ void k_gemm_wmma(const float* __restrict__ x,
                            const float* __restrict__ h,
                            const float* __restrict__ W,
                            float* __restrict__ xw) {
  int gwave = (blockIdx.x * blockDim.x + threadIdx.x) >> 5;
  int mt = gwave >> 4;                 // 0..4999  (M tile)
  int nt = gwave & 15;                 // 0..15    (N tile)
  int lane = threadIdx.x & 31;
  int lrow = lane & 15;
  int kq   = (lane >> 4) << 1;         // lanes 0-15 -> K {0,1}; lanes 16-31 -> K {2,3}

  long long row = (long long)mt * 16 + lrow;     // A: lane&15 = M
  const float* xrow = x + row * kCIN;
  const float* hrow = h + row * kH;
  int col = nt * 16 + lrow;                      // B: lane&15 = N

  v8f acc = {};
#pragma unroll
  for (int k = 0; k < kK; k += 4) {
    int ka = k + kq;                             // even; ka and ka+1 on same side of concat
    const float* abase = (ka < kCIN) ? (xrow + ka) : (hrow + (ka - kCIN));
    v2f a;  a.x = abase[0];             a.y = abase[1];                 // A VGPR0=K=ka, VGPR1=K=ka+1
    v2f bm; bm.x = W[(long long)ka * kF + col];
            bm.y = W[(long long)(ka + 1) * kF + col];                   // B VGPR0=K=ka, VGPR1=K=ka+1
    acc = __builtin_amdgcn_wmma_f32_16x16x4_f32(
        /*neg_a=*/false, a, /*neg_b=*/false, bm,
        /*c_mod=*/(short)0, acc, /*reuse_a=*/false, /*reuse_b=*/false);
  }

  // D layout: VGPR v -> M = v (lanes 0-15) or M = v+8 (lanes 16-31), N = lane&15
  long long drow = (long long)mt * 16 + ((lane >> 4) << 3);
#pragma unroll
  for (int v = 0; v < 8; ++v) {
    xw[(drow + v) * kF + col] = acc[v];
  }
}

// ---------- edge scatter: agg[b,dst,:] += xw[b,src,:] * norm(e) ----------
__global__ void k_edge_scatter(const long long* __restrict__ ei,
                               const float* __restrict__ ew,
                               const float* __restrict__ dis,
                               const float* __restrict__ xw,
                               float* __restrict__ agg) {
  int e = blockIdx.x;        // one block per edge
  int t = threadIdx.x;       // 0..255 feature column
  long long s = ei[e];
  long long d = ei[(long long)kE + e];
  float nrm = dis[s] * ew[e] * dis[d];
  const float* sp = xw + s * (long long)kF + t;
  float* dp = agg + d * (long long)kF + t;
  constexpr long long stride = (long long)kN * kF;
#pragma unroll
  for (int b = 0; b < kB; ++b) {
    atomicAdd(dp + (long long)b * stride, sp[(long long)b * stride] * nrm);
  }
}

// ---------- self-loop: agg[b,n,:] += xw[b,n,:] * dis[n]^2 (no atomics needed) ----------
__global__ void k_self_loop(const float* __restrict__ dis,
                            const float* __restrict__ xw,
                            float* __restrict__ agg) {
  long long i = (long long)blockIdx.x * blockDim.x + threadIdx.x;
  if (i < kBNF) {
    long long n = (i >> 8) % kN;   // kF == 256
    float di = dis[n];
    agg[i] += xw[i] * di * di;
  }
}

// ---------- LSTM gates ----------
static __device__ __forceinline__ float sigmoidf_(float v) {
  return 1.0f / (1.0f + __expf(-v));
}

__global__ void k_gates(const float* __restrict__ agg,
                        const float* __restrict__ c,
                        const float* __restrict__ bias,
                        float* __restrict__ out) {
  long long i = (long long)blockIdx.x * blockDim.x + threadIdx.x;
  if (i < kBNH) {
    long long bn = i >> 6;              // kH == 64
    int hc = (int)(i & 63);
    const float* a = agg + bn * kF;
    float ig = sigmoidf_(a[hc]          + bias[hc]);
    float fg = sigmoidf_(a[kH + hc]     + bias[kH + hc]);
    float og = sigmoidf_(a[2 * kH + hc] + bias[2 * kH + hc]);
    float gg = tanhf    (a[3 * kH + hc] + bias[3 * kH + hc]);
    float cn = fg * c[i] + ig * gg;
    float hn = og * tanhf(cn);
    out[i]        = hn;   // h_next
    out[kBNH + i] = cn;   // c_next
  }
}

extern "C" void kernel_launch(void* const* d_in, const int* in_sizes, int n_in,
                              void* d_out, int out_size, void* d_ws, size_t ws_size,
                              hipStream_t stream) {
  const float*     x    = (const float*)d_in[0];
  const float*     h    = (const float*)d_in[1];
  const float*     c    = (const float*)d_in[2];
  const long long* ei   = (const long long*)d_in[3];  // int64 [2, E]
  const float*     ew   = (const float*)d_in[4];
  const float*     W    = (const float*)d_in[5];      // [128, 256] row-major
  const float*     bias = (const float*)d_in[6];
  float*           out  = (float*)d_out;

  // Workspace layout: deg/dis (kN, padded), xw [B,N,256], agg [B,N,256]
  float* deg = (float*)d_ws;
  float* xw  = deg + ((kN + 63) & ~63);
  float* agg = xw + kBNF;

  k_deg_init  <<<(kN + 255) / 256, 256, 0, stream>>>(deg);
  k_deg_accum <<<(kE + 255) / 256, 256, 0, stream>>>(ei, ew, deg);
  k_deg_rsqrt <<<(kN + 255) / 256, 256, 0, stream>>>(deg);

  k_zero<<<(int)((kBNF + 255) / 256), 256, 0, stream>>>(agg, kBNF);

  // 5000 M-tiles * 16 N-tiles = 80000 waves; 8 waves per 256-thread block
  k_gemm_wmma<<<10000, 256, 0, stream>>>(x, h, W, xw);

  k_edge_scatter<<<kE, 256, 0, stream>>>(ei, ew, deg, xw, agg);
  k_self_loop<<<(int)((kBNF + 255) / 256), 256, 0, stream>>>(deg, xw, agg);
  k_gates<<<(int)((kBNH + 255) / 256), 256, 0, stream>>>(agg, c, bias, out);
}